// EdgeConvBlock_75093208203678
// MI455X (gfx1250) — compile-verified
//
#include <hip/hip_runtime.h>
#include <hip/hip_bf16.h>

typedef __attribute__((ext_vector_type(2))) float v2f;
typedef __attribute__((ext_vector_type(8))) float v8f;

#define KNN   20
#define NPTS  4096
#define NB    8
#define NCH   3
#define NOUT  64
#define EPS_F 1e-5f
#define SLOPE 0.2f
#define NEG_INF (-3.402823e38f)

// ---------------------------------------------------------------------------
// Kernel 1: fused pairwise-distance + top-K(20) per point.
// One block = 256 points of one batch. Whole batch (3x4096 floats + norms)
// staged in LDS (64 KB of the 320 KB WGP pool). Each thread scans all 4096
// columns, maintaining a sorted top-20 (value,index) list in registers,
// guarded by one compare per candidate. pd = 2*dot - |xi|^2 - |xj|^2.
// ---------------------------------------------------------------------------
__global__ __launch_bounds__(256) void topk_kernel(const float* __restrict__ x,
                                                   int* __restrict__ idx_out)
{
    __shared__ float xs0[NPTS];
    __shared__ float xs1[NPTS];
    __shared__ float xs2[NPTS];
    __shared__ float xxs[NPTS];

    const int blocks_per_batch = NPTS / 256;
    const int b     = blockIdx.x / blocks_per_batch;
    const int nbase = (blockIdx.x % blocks_per_batch) * 256;
    const float* xb = x + (size_t)b * NCH * NPTS;

    for (int i = threadIdx.x; i < NPTS; i += 256) {
        float a0 = xb[i];
        float a1 = xb[NPTS + i];
        float a2 = xb[2 * NPTS + i];
        xs0[i] = a0; xs1[i] = a1; xs2[i] = a2;
        xxs[i] = a0 * a0 + a1 * a1 + a2 * a2;
    }
    __syncthreads();

    const int n = nbase + threadIdx.x;
    const float xi0 = xs0[n], xi1 = xs1[n], xi2 = xs2[n], xxi = xxs[n];

    float bestv[KNN];
    int   besti[KNN];
#pragma unroll
    for (int k = 0; k < KNN; ++k) { bestv[k] = NEG_INF; besti[k] = 0; }

    for (int j = 0; j < NPTS; ++j) {
        // uniform j across the wave -> LDS broadcast reads, no bank conflicts
        float d = 2.0f * (xi0 * xs0[j] + xi1 * xs1[j] + xi2 * xs2[j]) - xxi - xxs[j];
        if (d > bestv[KNN - 1]) {          // rare path (~166 / 4096)
            float v = d; int id = j;
#pragma unroll
            for (int k = 0; k < KNN; ++k) {
                if (v > bestv[k]) {
                    float tv = bestv[k]; int ti = besti[k];
                    bestv[k] = v; besti[k] = id;
                    v = tv; id = ti;
                }
            }
        }
    }

    int* op = idx_out + ((size_t)b * NPTS + n) * KNN;
#pragma unroll
    for (int k = 0; k < KNN; ++k) op[k] = besti[k];
}

// ---------------------------------------------------------------------------
// Kernel 2: gather + edge GEMM (V_WMMA_F32_16X16X4_F32) + BN + LeakyReLU +
// max over K. One wave (32 lanes) per point.
//   A tiles : 2 x (16 edges x 4 chans)  -- edges 0..19 padded to 32 rows
//   B tiles : 4 x (4 chans x 16 outs)   -- W^T, channels 6..7 zero-padded
//   2 inner-K steps cover the 8 (padded) channels -> 16 WMMAs / point.
// A layout (ISA 7.12.2): lanes 0-15 carry K=0,1; lanes 16-31 carry K=2,3
// for the same M rows. C layout: VGPR r holds M=r (lanes 0-15) / M=r+8
// (lanes 16-31); M-reduction = max over 8 VGPRs then __shfl_xor(16).
// Padded edge rows are excluded from the max with -inf masking.
// ---------------------------------------------------------------------------
__global__ __launch_bounds__(256) void edgeconv_kernel(
    const float* __restrict__ x, const float* __restrict__ W,
    const float* __restrict__ gamma, const float* __restrict__ beta,
    const float* __restrict__ mean,  const float* __restrict__ var,
    const int* __restrict__ idx, float* __restrict__ out)
{
    const int wave = (int)((blockIdx.x * blockDim.x + threadIdx.x) >> 5);
    const int lane = threadIdx.x & 31;
    const int b = wave / NPTS;
    const int n = wave % NPTS;

    const float* xb = x + (size_t)b * NCH * NPTS;
    const float xi0 = xb[n], xi1 = xb[NPTS + n], xi2 = xb[2 * NPTS + n];

    const int  mrow = lane & 15;
    const bool hi   = lane >= 16;          // lanes 16-31 supply K = 2,3

    // ---- B tiles: B[c][o] = W[o][c]; btile[kk][ct], o = ct*16 + mrow ----
    v2f btile[2][4];
#pragma unroll
    for (int kk = 0; kk < 2; ++kk) {
#pragma unroll
        for (int ct = 0; ct < 4; ++ct) {
            int o  = ct * 16 + mrow;
            int c0 = kk * 4 + (hi ? 2 : 0);
            float w0 = (c0     < 2 * NCH) ? W[o * (2 * NCH) + c0]     : 0.0f;
            float w1 = (c0 + 1 < 2 * NCH) ? W[o * (2 * NCH) + c0 + 1] : 0.0f;
            v2f bb; bb.x = w0; bb.y = w1;
            btile[kk][ct] = bb;
        }
    }

    v8f acc[2][4];
#pragma unroll
    for (int m = 0; m < 2; ++m)
#pragma unroll
        for (int ct = 0; ct < 4; ++ct)
            acc[m][ct] = (v8f){0.f, 0.f, 0.f, 0.f, 0.f, 0.f, 0.f, 0.f};

    const int* idxp = idx + ((size_t)b * NPTS + n) * KNN;

#pragma unroll
    for (int m = 0; m < 2; ++m) {
        int  e     = m * 16 + mrow;
        bool valid = (e < KNN);
        int  e2    = valid ? e : (KNN - 1);
        int  j     = idxp[e2];
        float xj0 = xb[j], xj1 = xb[NPTS + j], xj2 = xb[2 * NPTS + j];

        // edge feature channels: 0..2 = xj - xi, 3..5 = xi, 6..7 = pad
        float ch0 = valid ? (xj0 - xi0) : 0.0f;
        float ch1 = valid ? (xj1 - xi1) : 0.0f;
        float ch2 = valid ? (xj2 - xi2) : 0.0f;
        float ch3 = valid ? xi0 : 0.0f;
        float ch4 = valid ? xi1 : 0.0f;
        float ch5 = valid ? xi2 : 0.0f;

        v2f a0, a1;                        // kk=0 -> {0,1}/{2,3}; kk=1 -> {4,5}/{6,7}
        a0.x = hi ? ch2 : ch0;  a0.y = hi ? ch3 : ch1;
        a1.x = hi ? 0.0f : ch4; a1.y = hi ? 0.0f : ch5;

#pragma unroll
        for (int ct = 0; ct < 4; ++ct) {
            acc[m][ct] = __builtin_amdgcn_wmma_f32_16x16x4_f32(
                false, a0, false, btile[0][ct], (short)0, acc[m][ct], false, false);
            acc[m][ct] = __builtin_amdgcn_wmma_f32_16x16x4_f32(
                false, a1, false, btile[1][ct], (short)0, acc[m][ct], false, false);
        }
    }

    // ---- BN affine + LeakyReLU (pre-max, as in reference), then max over K ----
#pragma unroll
    for (int ct = 0; ct < 4; ++ct) {
        int   o   = ct * 16 + mrow;                  // this lane's output channel
        float inv = gamma[o] * rsqrtf(var[o] + EPS_F);
        float b2  = beta[o] - mean[o] * inv;

        // tile 0: edges 0..15, all 8 VGPRs valid
        float m0 = NEG_INF;
#pragma unroll
        for (int r = 0; r < 8; ++r) {
            float y = acc[0][ct][r] * inv + b2;
            y = (y >= 0.0f) ? y : SLOPE * y;
            m0 = fmaxf(m0, y);
        }
        m0 = fmaxf(m0, __shfl_xor(m0, 16, 32));      // combine M=0..7 with M=8..15

        // tile 1: only M=0..3 (edges 16..19) valid, and only in lanes 0-15
        float m1 = NEG_INF;
#pragma unroll
        for (int r = 0; r < 4; ++r) {
            float y = acc[1][ct][r] * inv + b2;
            y = (y >= 0.0f) ? y : SLOPE * y;
            m1 = fmaxf(m1, y);
        }
        m1 = hi ? NEG_INF : m1;                      // mask padded half
        m1 = fmaxf(m1, __shfl_xor(m1, 16, 32));

        float res = fmaxf(m0, m1);
        if (!hi)
            out[((size_t)b * NOUT + o) * NPTS + n] = res;
    }
}

extern "C" void kernel_launch(void* const* d_in, const int* in_sizes, int n_in,
                              void* d_out, int out_size, void* d_ws, size_t ws_size,
                              hipStream_t stream) {
    const float* x     = (const float*)d_in[0];
    const float* W     = (const float*)d_in[1];
    const float* gamma = (const float*)d_in[2];
    const float* beta  = (const float*)d_in[3];
    const float* mean  = (const float*)d_in[4];
    const float* var   = (const float*)d_in[5];
    float* out = (float*)d_out;
    int*   idx = (int*)d_ws;                         // B*N*K ints = 2.62 MB

    topk_kernel<<<NB * (NPTS / 256), 256, 0, stream>>>(x, idx);
    edgeconv_kernel<<<(NB * NPTS) / 8, 256, 0, stream>>>(x, W, gamma, beta,
                                                         mean, var, idx, out);
}